// SingleStepTextDecoder_88502096101500
// MI455X (gfx1250) — compile-verified
//
#include <hip/hip_runtime.h>
#include <hip/hip_bf16.h>

#define LNUM 4
#define DIM 2048
#define NH 32
#define DH 64
#define FF 8192
#define CTXN 4096
#define VOCAB 32000
#define ASPLIT 4

typedef float v2f_ __attribute__((ext_vector_type(2)));
typedef float v8f_ __attribute__((ext_vector_type(8)));

// ---------------- LayerNorm (single block) ----------------
__global__ void layernorm_k(const float* __restrict__ x, const float* __restrict__ w,
                            const float* __restrict__ b, float* __restrict__ out, int n) {
    __shared__ float s1[256], s2[256];
    int tid = threadIdx.x;
    float sum = 0.f, sq = 0.f;
    for (int i = tid; i < n; i += 256) { float v = x[i]; sum += v; sq += v * v; }
    s1[tid] = sum; s2[tid] = sq; __syncthreads();
    for (int off = 128; off > 0; off >>= 1) {
        if (tid < off) { s1[tid] += s1[tid + off]; s2[tid] += s2[tid + off]; }
        __syncthreads();
    }
    float mu = s1[0] / n;
    float var = s2[0] / n - mu * mu;
    float inv = rsqrtf(var + 1e-5f);
    for (int i = tid; i < n; i += 256) out[i] = (x[i] - mu) * inv * w[i] + b[i];
}

// ------- WMMA split-K GEMV partial: part[sk*O+o] = sum_{d in slab} x[d]*W[d*O+o] -------
// Block = 256 threads = 8 waves; each wave -> 16 outputs. grid = (O/128, SK).
// A (16x4 f32) = x broadcast over rows; B (4x16 f32) = W tile. Row 0 of D holds result.
__global__ void gemv_wmma_part_k(const float* __restrict__ x, const float* __restrict__ W,
                                 float* __restrict__ part, int O, int slab) {
    extern __shared__ float sx[];
    const int sk = blockIdx.y;
    const int k0 = sk * slab;
    for (int i = threadIdx.x; i < slab; i += blockDim.x) sx[i] = x[k0 + i];
    __syncthreads();
    const int lane = threadIdx.x & 31;
    const int wave = threadIdx.x >> 5;
    const int o0 = blockIdx.x * 128 + wave * 16;
    const int lm = lane & 15;
    const int koff0 = (lane >= 16) ? 2 : 0;  // A/B VGPR0: K=0 (lo half) / K=2 (hi half)
    const int koff1 = koff0 + 1;             // A/B VGPR1: K=1 / K=3
    v8f_ c = {0.f, 0.f, 0.f, 0.f, 0.f, 0.f, 0.f, 0.f};
    const float* wp = W + (size_t)k0 * O + o0 + lm;
#pragma unroll 8
    for (int d0 = 0; d0 < slab; d0 += 4) {
        v2f_ a, b;
        a.x = sx[d0 + koff0];
        a.y = sx[d0 + koff1];
        b.x = wp[(size_t)(d0 + koff0) * O];
        b.y = wp[(size_t)(d0 + koff1) * O];
        c = __builtin_amdgcn_wmma_f32_16x16x4_f32(false, a, false, b, (short)0, c,
                                                  false, false);
    }
    if (lane < 16) part[(size_t)sk * O + o0 + lane] = c[0];
}

// ------- reduce partials + bias (+GELU) -------
__global__ void gemv_reduce_k(const float* __restrict__ part, const float* __restrict__ bias,
                              float* __restrict__ y, int O, int SK, int act) {
    int o = blockIdx.x * 256 + threadIdx.x;
    if (o >= O) return;
    float v = bias[o];
    for (int s = 0; s < SK; ++s) v += part[(size_t)s * O + o];
    if (act == 1) {  // tanh-approx GELU
        float t = 0.7978845608028654f * (v + 0.044715f * v * v * v);
        v = 0.5f * v * (1.0f + tanhf(t));
    }
    y[o] = v;
}

// ---------------- RoPE + qkv split ----------------
__global__ void rope_k(const float* __restrict__ qkv, const float* __restrict__ freqs,
                       const int* __restrict__ pos_id, float* __restrict__ qr,
                       float* __restrict__ kr, float* __restrict__ vv) {
    int idx = blockIdx.x * blockDim.x + threadIdx.x;  // 0..2047
    int h = idx >> 6, d = idx & 63;
    int pos = pos_id[0];
    const float* q = qkv;
    const float* k = qkv + DIM;
    float qo, ko;
    if (d < 32) {
        int i = d >> 1;
        float cs = freqs[(pos * 16 + i) * 2 + 0];
        float sn = freqs[(pos * 16 + i) * 2 + 1];
        float qxr = q[h * 64 + i], qxi = q[h * 64 + 16 + i];
        float kxr = k[h * 64 + i], kxi = k[h * 64 + 16 + i];
        if (d & 1) { qo = qxr * sn + qxi * cs; ko = kxr * sn + kxi * cs; }
        else       { qo = qxr * cs - qxi * sn; ko = kxr * cs - kxi * sn; }
    } else {
        qo = q[h * 64 + d];
        ko = k[h * 64 + d];
    }
    qr[idx] = qo; kr[idx] = ko; vv[idx] = qkv[2 * DIM + idx];
}

// ------- split attention: grid (NH, ASPLIT); partial = [max, sum, acc[64]] -------
__global__ void attn_part_k(const float* __restrict__ qr, const float* __restrict__ kr,
                            const float* __restrict__ vv, const float* __restrict__ kc,
                            const float* __restrict__ vc, const float* __restrict__ mask,
                            const int* __restrict__ pos_id, float* __restrict__ apart) {
    __shared__ float sc[(CTXN + ASPLIT - 1) / ASPLIT];
    __shared__ float sq[64];
    __shared__ float red[256];
    __shared__ float vred[4][64];
    const int h = blockIdx.x, sp = blockIdx.y, tid = threadIdx.x;
    const int pos = pos_id[0];
    const int total = pos + 1;
    const int chunk = (total + ASPLIT - 1) / ASPLIT;
    const int j0 = sp * chunk;
    const int j1 = (j0 + chunk < total) ? (j0 + chunk) : total;
    if (tid < 64) sq[tid] = qr[h * 64 + tid];
    __syncthreads();
    float lmax = -3.4e38f;
    for (int j = j0 + tid; j < j1; j += 256) {
        const float* krow = (j == pos) ? (kr + h * 64)
                                       : (kc + ((size_t)h * CTXN + j) * DH);
        float s = 0.f;
#pragma unroll
        for (int d = 0; d < 64; d += 4) {
            const float4 k4 = *reinterpret_cast<const float4*>(krow + d);
            s += sq[d] * k4.x + sq[d + 1] * k4.y + sq[d + 2] * k4.z + sq[d + 3] * k4.w;
        }
        s = s * 0.125f + mask[j];
        sc[j - j0] = s;
        lmax = fmaxf(lmax, s);
    }
    red[tid] = lmax; __syncthreads();
    for (int off = 128; off > 0; off >>= 1) {
        if (tid < off) red[tid] = fmaxf(red[tid], red[tid + off]);
        __syncthreads();
    }
    const float gmax = red[0];
    __syncthreads();
    float lsum = 0.f;
    for (int j = j0 + tid; j < j1; j += 256) {
        float e = expf(sc[j - j0] - gmax);
        sc[j - j0] = e;
        lsum += e;
    }
    red[tid] = lsum; __syncthreads();
    for (int off = 128; off > 0; off >>= 1) {
        if (tid < off) red[tid] += red[tid + off];
        __syncthreads();
    }
    const float gsum = red[0];
    const int jc = tid >> 6, d = tid & 63;
    float acc = 0.f;
    for (int j = j0 + jc; j < j1; j += 4) {
        const float* vrow = (j == pos) ? (vv + h * 64)
                                       : (vc + ((size_t)h * CTXN + j) * DH);
        acc += sc[j - j0] * vrow[d];
    }
    vred[jc][d] = acc; __syncthreads();
    float* outp = apart + ((size_t)h * ASPLIT + sp) * 66;
    if (tid < 64)
        outp[2 + tid] = vred[0][tid] + vred[1][tid] + vred[2][tid] + vred[3][tid];
    if (tid == 0) { outp[0] = gmax; outp[1] = gsum; }
}

// ------- merge attention partials: grid NH, 64 threads -------
__global__ void attn_merge_k(const float* __restrict__ apart, float* __restrict__ out) {
    const int h = blockIdx.x, tid = threadIdx.x;  // tid = d
    const float* p = apart + (size_t)h * ASPLIT * 66;
    float M = -3.4e38f;
    for (int i = 0; i < ASPLIT; ++i) M = fmaxf(M, p[i * 66]);
    float denom = 0.f, num = 0.f;
    for (int i = 0; i < ASPLIT; ++i) {
        float w = expf(p[i * 66] - M);
        denom += w * p[i * 66 + 1];
        num += w * p[i * 66 + 2 + tid];
    }
    out[h * 64 + tid] = num / denom;
}

// ---------------- small helpers ----------------
__global__ void copy_k(const float* __restrict__ src, float* __restrict__ dst, int n) {
    int i = blockIdx.x * blockDim.x + threadIdx.x;
    if (i < n) dst[i] = src[i];
}
__global__ void addres_k(float* __restrict__ x, const float* __restrict__ a,
                         const float* __restrict__ m, int n) {
    int i = blockIdx.x * blockDim.x + threadIdx.x;
    if (i < n) x[i] = x[i] + a[i] + m[i];
}

// host-side helper: split-K GEMV launch
static void gemv(const float* x, const float* W, const float* bias, float* y,
                 float* part, int D, int O, int SK, int act, hipStream_t stream) {
    int slab = D / SK;
    dim3 grid(O / 128, SK);
    gemv_wmma_part_k<<<grid, 256, slab * sizeof(float), stream>>>(x, W, part, O, slab);
    gemv_reduce_k<<<(O + 255) / 256, 256, 0, stream>>>(part, bias, y, O, SK, act);
}

extern "C" void kernel_launch(void* const* d_in, const int* in_sizes, int n_in,
                              void* d_out, int out_size, void* d_ws, size_t ws_size,
                              hipStream_t stream) {
    const float* token_embed = (const float*)d_in[0];
    const int*   pos_id      = (const int*)d_in[1];
    const float* attn_mask   = (const float*)d_in[2];
    const float* freqs       = (const float*)d_in[3];
    const float* k_cache     = (const float*)d_in[4];
    const float* v_cache     = (const float*)d_in[5];
    const float* ln_w        = (const float*)d_in[6];
    const float* ln_b        = (const float*)d_in[7];
    const float* qkv_w       = (const float*)d_in[8];
    const float* qkv_b       = (const float*)d_in[9];
    const float* proj_w      = (const float*)d_in[10];
    const float* proj_b      = (const float*)d_in[11];
    const float* fc1_w       = (const float*)d_in[12];
    const float* fc1_b       = (const float*)d_in[13];
    const float* fc2_w       = (const float*)d_in[14];
    const float* fc2_b       = (const float*)d_in[15];
    const float* post_ln_w   = (const float*)d_in[16];
    const float* post_ln_b   = (const float*)d_in[17];
    const float* lm_head_w   = (const float*)d_in[18];
    const float* lm_head_b   = (const float*)d_in[19];

    float* ws  = (float*)d_ws;
    float* X     = ws + 0;        // 2048 residual
    float* H     = ws + 2048;     // 2048 LN output
    float* QKV   = ws + 4096;     // 6144
    float* QR    = ws + 10240;    // 2048
    float* KR    = ws + 12288;    // 2048
    float* VV    = ws + 14336;    // 2048
    float* AT    = ws + 16384;    // 2048 attention out
    float* PA    = ws + 18432;    // 2048 proj out
    float* M1    = ws + 20480;    // 8192 fc1/gelu out
    float* M2    = ws + 28672;    // 2048 fc2 out
    float* APART = ws + 30720;    // NH*ASPLIT*66 = 8448 attention partials
    float* GPART = ws + 40960;    // up to 4*32000 = 128000 gemv partials
    float* logits = (float*)d_out;

    const size_t cache_stride = (size_t)NH * CTXN * DH;

    copy_k<<<8, 256, 0, stream>>>(token_embed, X, DIM);

    for (int l = 0; l < LNUM; ++l) {
        layernorm_k<<<1, 256, 0, stream>>>(X, ln_w + (size_t)l * DIM,
                                           ln_b + (size_t)l * DIM, H, DIM);
        gemv(H, qkv_w + (size_t)l * DIM * 3 * DIM, qkv_b + (size_t)l * 3 * DIM, QKV,
             GPART, DIM, 3 * DIM, /*SK=*/8, 0, stream);
        rope_k<<<8, 256, 0, stream>>>(QKV, freqs, pos_id, QR, KR, VV);
        attn_part_k<<<dim3(NH, ASPLIT), 256, 0, stream>>>(
            QR, KR, VV, k_cache + (size_t)l * cache_stride,
            v_cache + (size_t)l * cache_stride, attn_mask, pos_id, APART);
        attn_merge_k<<<NH, 64, 0, stream>>>(APART, AT);
        gemv(AT, proj_w + (size_t)l * DIM * DIM, proj_b + (size_t)l * DIM, PA,
             GPART, DIM, DIM, /*SK=*/16, 0, stream);
        gemv(H, fc1_w + (size_t)l * DIM * FF, fc1_b + (size_t)l * FF, M1,
             GPART, DIM, FF, /*SK=*/8, 1, stream);
        gemv(M1, fc2_w + (size_t)l * FF * DIM, fc2_b + (size_t)l * DIM, M2,
             GPART, FF, DIM, /*SK=*/32, 0, stream);
        addres_k<<<8, 256, 0, stream>>>(X, PA, M2, DIM);
    }

    layernorm_k<<<1, 256, 0, stream>>>(X, post_ln_w, post_ln_b, H, DIM);
    gemv(H, lm_head_w, lm_head_b, logits, GPART, DIM, VOCAB, /*SK=*/4, 0, stream);
    copy_k<<<8, 256, 0, stream>>>(X, logits + VOCAB, DIM);
}